// SenShift_Feat_8993661518510
// MI455X (gfx1250) — compile-verified
//
#include <hip/hip_runtime.h>
#include <hip/hip_bf16.h>

// ---------------------------------------------------------------------------
// SenShift pairwise-logit kernel for MI455X (gfx1250, wave32).
//
// reference:
//   proj_i = embeds      @ w_i.T      (n x 256) . (256 x 2)
//   proj_j = embeds_temp @ w_j.T      (n x 256) . (256 x 2)
//   out    = proj_i[idx_i] + proj_j[idx_j] + bias   (P x 2)
//
// Roofline: ~0.13 GFLOP vs ~68 MB traffic -> ~3 us @ 23.3 TB/s HBM; fully
// memory-bound. Strategy:
//   K1: per-wave 16-row tile, two V_WMMA_F32_16X16X4_F32 accumulation chains
//       (full f32 precision; no mask needed on B -- WMMA columns are
//       independent and only N=0,1 are ever read from C, so columns 2..15
//       just accumulate harmless duplicates of weight row n&1). Each
//       embedding row is read from HBM exactly once.
//   K2: coalesced float2 gather-add over the 512K pairs; proj arrays
//       (512 KB) are L2-resident.
// ---------------------------------------------------------------------------

typedef float v2f __attribute__((ext_vector_type(2)));
typedef float v8f __attribute__((ext_vector_type(8)));

#define HIDDEN 256

__global__ __launch_bounds__(256)
void senshift_proj_wmma(const float* __restrict__ embeds,
                        const float* __restrict__ embeds_temp,
                        const float* __restrict__ weight,   // (2, 512)
                        float* __restrict__ proj_i,         // (n, 2)
                        float* __restrict__ proj_j,         // (n, 2)
                        int nrows) {
    const int lane  = threadIdx.x & 31;
    const int wave  = threadIdx.x >> 5;
    const int tile  = blockIdx.x * 8 + wave;   // one 16-row tile per wave
    const int r0    = tile * 16;
    if (r0 + 16 > nrows) return;               // wave-uniform; EXEC stays all-1s

    // A-operand (16x4 f32): lanes 0-15 hold K={0,1} for row M=lane,
    //                       lanes 16-31 hold K={2,3} for row M=lane-16.
    const int m     = lane & 15;               // row-in-tile for A; column N for B/C
    const int khalf = lane >> 4;               // selects K pair {0,1} vs {2,3}

    const float* rowA = embeds      + (size_t)(r0 + m) * HIDDEN + 2 * khalf;
    const float* rowB = embeds_temp + (size_t)(r0 + m) * HIDDEN + 2 * khalf;

    // B-operand (4x16 f32): lane's N = m. Only N=0,1 are ever read from the
    // accumulator; lanes N>=2 load weight row (m&1) unmasked -- their columns
    // accumulate finite duplicates that are simply never stored.
    const float* wrow = weight + (size_t)(m & 1) * (2 * HIDDEN) + 2 * khalf;

    v8f c1 = {};   // embeds      . w_i
    v8f c2 = {};   // embeds_temp . w_j

    #pragma unroll 4
    for (int k0 = 0; k0 < HIDDEN; k0 += 4) {
        v2f a1 = { rowA[k0],          rowA[k0 + 1] };
        v2f a2 = { rowB[k0],          rowB[k0 + 1] };
        v2f b1 = { wrow[k0],          wrow[k0 + 1] };
        v2f b2 = { wrow[k0 + HIDDEN], wrow[k0 + 1 + HIDDEN] };

        c1 = __builtin_amdgcn_wmma_f32_16x16x4_f32(
                 /*neg_a=*/false, a1, /*neg_b=*/false, b1,
                 /*c_mod=*/(short)0, c1, /*reuse_a=*/false, /*reuse_b=*/false);
        c2 = __builtin_amdgcn_wmma_f32_16x16x4_f32(
                 /*neg_a=*/false, a2, /*neg_b=*/false, b2,
                 /*c_mod=*/(short)0, c2, /*reuse_a=*/false, /*reuse_b=*/false);
    }

    // C/D layout (16x16 f32): VGPR v, lanes 0-15 -> (M=v,     N=lane),
    //                                  lanes 16-31 -> (M=v+8, N=lane-16).
    // Only columns N=0,1 are meaningful -> 4 lanes store 8 floats each.
    if (m < 2) {
        const int baseRow = r0 + khalf * 8;
        #pragma unroll
        for (int v = 0; v < 8; ++v) {
            proj_i[(size_t)(baseRow + v) * 2 + m] = c1[v];
            proj_j[(size_t)(baseRow + v) * 2 + m] = c2[v];
        }
    }
}

__global__ __launch_bounds__(256)
void senshift_gather_add(const float* __restrict__ proj_i,
                         const float* __restrict__ proj_j,
                         const int*   __restrict__ idx_i,
                         const int*   __restrict__ idx_j,
                         const float* __restrict__ bias,
                         float* __restrict__ out,
                         int npairs) {
    int p = blockIdx.x * blockDim.x + threadIdx.x;
    if (p >= npairs) return;

    const int i = idx_i[p];
    const int j = idx_j[p];

    const float2 pi = ((const float2*)proj_i)[i];
    const float2 pj = ((const float2*)proj_j)[j];

    float2 o;
    o.x = pi.x + pj.x + bias[0];
    o.y = pi.y + pj.y + bias[1];
    ((float2*)out)[p] = o;
}

extern "C" void kernel_launch(void* const* d_in, const int* in_sizes, int n_in,
                              void* d_out, int out_size, void* d_ws, size_t ws_size,
                              hipStream_t stream) {
    const float* embeds      = (const float*)d_in[0];
    const float* embeds_temp = (const float*)d_in[1];
    const float* weight      = (const float*)d_in[2];   // (2, 512)
    const float* bias        = (const float*)d_in[3];   // (2,)
    const int*   idx_i       = (const int*)d_in[4];
    const int*   idx_j       = (const int*)d_in[5];

    const int nrows  = in_sizes[0] / HIDDEN;            // 32768
    const int npairs = in_sizes[4];                     // 512512

    float* proj_i = (float*)d_ws;
    float* proj_j = proj_i + (size_t)nrows * 2;

    // K1: one wave per 16-row tile, 8 waves (256 threads) per block.
    const int ntiles  = (nrows + 15) / 16;
    const int nblocks = (ntiles + 7) / 8;
    senshift_proj_wmma<<<nblocks, 256, 0, stream>>>(
        embeds, embeds_temp, weight, proj_i, proj_j, nrows);

    // K2: pair gather + bias.
    const int gblocks = (npairs + 255) / 256;
    senshift_gather_add<<<gblocks, 256, 0, stream>>>(
        proj_i, proj_j, idx_i, idx_j, bias, (float*)d_out, npairs);

    (void)n_in; (void)out_size; (void)ws_size;
}